// MLGraphConstruction_35390530519953
// MI455X (gfx1250) — compile-verified
//
#include <hip/hip_runtime.h>
#include <stdint.h>

#define N_PTS 12288
#define DIMS  14
#define KNN   256

typedef __attribute__((ext_vector_type(2))) float v2f;
typedef __attribute__((ext_vector_type(8))) float v8f;

// ---------------------------------------------------------------------------
// Kernel 1: squared norms sq[i] = sum_d x[i][d]^2
// ---------------------------------------------------------------------------
__global__ void sq_kernel(const float* __restrict__ x, float* __restrict__ sq) {
  int i = blockIdx.x * blockDim.x + threadIdx.x;
  if (i < N_PTS) {
    float s = 0.0f;
#pragma unroll
    for (int d = 0; d < DIMS; ++d) {
      float v = x[i * DIMS + d];
      s += v * v;
    }
    sq[i] = s;
  }
}

// ---------------------------------------------------------------------------
// Kernel 2: WMMA-tiled distance computation + exact top-256 selection.
// One wave (32 lanes) owns a 16-row block. d2 tiles are produced with
// V_WMMA_F32_16X16X4_F32 (K padded 14->16, 4 accumulating steps), then
// lanes 0-15 maintain per-row 256-entry max-heaps in LDS.
// Per-wave LDS: 16*256 u64 heap (32KB) + 16x17 f32 tile scratch.
// ---------------------------------------------------------------------------
#define WAVE_U64 (16 * 256 + 136)  // 4096 heap entries + 272 floats (136 u64)

__launch_bounds__(128, 1)
__global__ void knn_kernel(const float* __restrict__ x,
                           const float* __restrict__ sq,
                           float* __restrict__ topk_d2,
                           int* __restrict__ topk_idx) {
  extern __shared__ unsigned long long smem[];
  const int lane = threadIdx.x & 31;
  const int wave = threadIdx.x >> 5;                 // 0..3
  const int rowbase = (blockIdx.x * 4 + wave) * 16;  // 768 row blocks total

  unsigned long long* heap = smem + (size_t)wave * WAVE_U64;
  float* tile = (float*)(heap + 16 * 256);

  // init heaps to +inf keys
  const unsigned long long INFKEY = 0x7F800000ull << 32;
  for (int t = lane; t < 16 * 256; t += 32) heap[t] = INFKEY;
  asm volatile("s_wait_dscnt 0x0" ::: "memory");

  const int lmod = lane & 15;
  const int lhi  = lane >> 4;

  // A fragment (rows block), resident for the whole sweep.
  // 32-bit A 16x4 layout: lanes 0-15 hold K=kk+0,kk+1; lanes 16-31 K=kk+2,kk+3.
  const int arow = rowbase + lmod;
  v2f afrag[4];
#pragma unroll
  for (int s = 0; s < 4; ++s) {
    int k0 = s * 4 + lhi * 2;
    float ax = (k0     < DIMS) ? x[arow * DIMS + k0]     : 0.0f;
    float ay = (k0 + 1 < DIMS) ? x[arow * DIMS + k0 + 1] : 0.0f;
    afrag[s].x = ax;
    afrag[s].y = ay;
  }
  float sqr[8];
#pragma unroll
  for (int v = 0; v < 8; ++v) sqr[v] = sq[rowbase + v + lhi * 8];

  for (int cb = 0; cb < N_PTS; cb += 16) {
    const int col = cb + lmod;
    if (cb + 16 < N_PTS)
      __builtin_prefetch(&x[(size_t)(cb + 16 + lmod) * DIMS], 0, 3);

    // B fragment (cols block of x^T): mirror of A layout, col = lane%16.
    v8f acc = {};
#pragma unroll
    for (int s = 0; s < 4; ++s) {
      int k0 = s * 4 + lhi * 2;
      v2f bfrag;
      bfrag.x = (k0     < DIMS) ? x[col * DIMS + k0]     : 0.0f;
      bfrag.y = (k0 + 1 < DIMS) ? x[col * DIMS + k0 + 1] : 0.0f;
      acc = __builtin_amdgcn_wmma_f32_16x16x4_f32(
          /*neg_a=*/false, afrag[s], /*neg_b=*/false, bfrag,
          /*c_mod=*/(short)0, acc, /*reuse_a=*/false, /*reuse_b=*/false);
    }

    const float sqc = sq[col];
    // d2 = |xi|^2 + |xj|^2 - 2*dot ; scatter 16x16 tile to LDS (row-major, +1 pad)
#pragma unroll
    for (int v = 0; v < 8; ++v) {
      float d2 = sqr[v] + sqc - 2.0f * acc[v];
      d2 = fmaxf(d2, 0.0f);
      tile[(v + lhi * 8) * 17 + lmod] = d2;
    }
    asm volatile("s_wait_dscnt 0x0" ::: "memory");

    // selection: lane r (0..15) owns the max-heap for row rowbase+r
    if (lane < 16) {
      unsigned long long* h = heap + lane * 256;
      unsigned long long root = h[0];
      const int grow = rowbase + lane;
      for (int c = 0; c < 16; ++c) {
        const int j = cb + c;
        if (j == grow) continue;  // exclude self-loop
        const float d2 = tile[lane * 17 + c];
        const unsigned long long key =
            ((unsigned long long)__float_as_uint(d2) << 32) | (unsigned)j;
        if (key < root) {
          unsigned ii = 0;
          for (;;) {
            unsigned cc = 2 * ii + 1;
            if (cc >= 256u) break;
            unsigned long long hc = h[cc];
            if (cc + 1 < 256u) {
              unsigned long long h2 = h[cc + 1];
              if (h2 > hc) { hc = h2; ++cc; }
            }
            if (hc <= key) break;
            h[ii] = hc;
            ii = cc;
          }
          h[ii] = key;
          root = h[0];
        }
      }
    }
    asm volatile("s_wait_dscnt 0x0" ::: "memory");
  }

  // in-place heapsort -> ascending d2 (matches top_k ordering)
  if (lane < 16) {
    unsigned long long* h = heap + lane * 256;
    for (int n = 255; n > 0; --n) {
      const unsigned long long last = h[n];
      h[n] = h[0];
      unsigned ii = 0;
      for (;;) {
        unsigned cc = 2 * ii + 1;
        if (cc >= (unsigned)n) break;
        unsigned long long hc = h[cc];
        if (cc + 1 < (unsigned)n) {
          unsigned long long h2 = h[cc + 1];
          if (h2 > hc) { hc = h2; ++cc; }
        }
        if (hc <= last) break;
        h[ii] = hc;
        ii = cc;
      }
      h[ii] = last;
    }
  }
  asm volatile("s_wait_dscnt 0x0" ::: "memory");

  // flush sorted results to workspace
  for (int t = lane; t < 16 * KNN; t += 32) {
    const unsigned long long key = heap[t];
    const int r = t >> 8, k = t & 255;
    const size_t gi = (size_t)(rowbase + r) * KNN + k;
    topk_d2[gi]  = __uint_as_float((unsigned)(key >> 32));
    topk_idx[gi] = (int)(unsigned)(key & 0xFFFFFFFFu);
  }
}

// ---------------------------------------------------------------------------
// Kernel 3a: pass-through copy of x into output slot 0
// ---------------------------------------------------------------------------
__global__ void copy_x_kernel(const float* __restrict__ x, float* __restrict__ out) {
  int i = blockIdx.x * blockDim.x + threadIdx.x;
  if (i < N_PTS * DIMS) out[i] = x[i];
}

// ---------------------------------------------------------------------------
// Kernel 3b: assemble edge_index / y / edge_features / mask (memory-bound)
// ---------------------------------------------------------------------------
__global__ void assemble_kernel(const float* __restrict__ x,
                                const int* __restrict__ pid,
                                const float* __restrict__ topk_d2,
                                const int* __restrict__ topk_idx,
                                float* __restrict__ out) {
  const long long E = (long long)N_PTS * KNN;
  const long long e = (long long)blockIdx.x * blockDim.x + threadIdx.x;
  if (e >= E) return;

  const int i = (int)(e >> 8);  // query (dst), K == 256
  const int j = topk_idx[e];    // neighbor (src)
  const float d2 = topk_d2[e];
  const float dist = sqrtf(d2);
  const bool m = dist < 1.0f;   // MAX_RADIUS
  const float fm = m ? 1.0f : 0.0f;

  const long long OFF_EI = (long long)N_PTS * DIMS;  // after x
  const long long OFF_Y  = OFF_EI + 2 * E;
  const long long OFF_F  = OFF_Y + E;
  const long long OFF_M  = OFF_F + E * 2 * DIMS;

  out[OFF_EI + e]     = m ? (float)j : 0.0f;  // edge_index[0]
  out[OFF_EI + E + e] = m ? (float)i : 0.0f;  // edge_index[1]
  out[OFF_Y + e]      = (m && (pid[j] == pid[i])) ? 1.0f : 0.0f;
  out[OFF_M + e]      = fm;

  float fbuf[2 * DIMS];
#pragma unroll
  for (int d = 0; d < DIMS; ++d) {
    const float xs = x[(size_t)j * DIMS + d];
    const float xd = x[(size_t)i * DIMS + d];
    fbuf[d]        = (xs - xd) * fm;
    fbuf[DIMS + d] = (xs + xd) * fm;
  }
  float* fout = out + OFF_F + e * (2 * DIMS);
#pragma unroll
  for (int d = 0; d < 2 * DIMS; ++d) fout[d] = fbuf[d];
}

// ---------------------------------------------------------------------------
extern "C" void kernel_launch(void* const* d_in, const int* in_sizes, int n_in,
                              void* d_out, int out_size, void* d_ws, size_t ws_size,
                              hipStream_t stream) {
  const float* x  = (const float*)d_in[0];
  const int* pid  = (const int*)d_in[1];
  float* out      = (float*)d_out;

  // workspace layout: sq[N] | topk_d2[N*K] | topk_idx[N*K]  (~25.2 MB)
  float* sq       = (float*)d_ws;
  float* topk_d2  = sq + N_PTS;
  int*   topk_idx = (int*)(topk_d2 + (size_t)N_PTS * KNN);

  sq_kernel<<<(N_PTS + 255) / 256, 256, 0, stream>>>(x, sq);

  const size_t lds_bytes = (size_t)4 * WAVE_U64 * sizeof(unsigned long long);
  knn_kernel<<<N_PTS / 16 / 4, 128, lds_bytes, stream>>>(x, sq, topk_d2, topk_idx);

  copy_x_kernel<<<(N_PTS * DIMS + 255) / 256, 256, 0, stream>>>(x, out);

  const long long E = (long long)N_PTS * KNN;
  assemble_kernel<<<(unsigned)((E + 255) / 256), 256, 0, stream>>>(
      x, pid, topk_d2, topk_idx, out);
}